// Update_hgt_80960133529950
// MI455X (gfx1250) — compile-verified
//
#include <hip/hip_runtime.h>
#include <math.h>

// ---------------------------------------------------------------------------
// HGT node update: type-bucketed GEMM (msg @ wa[type]) + bias + sigmoid-gated
// skip blend + LayerNorm.  fp32 WMMA (v_wmma_f32_16x16x4_f32) keeps reference
// fp32 numerics while using the CDNA5 matrix pipe.
//
// Roofline: 17.2 GFLOP / 384 MB HBM -> ~45 FLOP/B; fp32 matrix path needed to
// reach the 23.3 TB/s memory floor (~16.5 us). wa (2 MB) is L2-resident and
// register-cached per wave (B slab reused across all node tiles of the type).
//
// Round-2 change: 4 independent WMMA accumulation chains (even/odd K split)
// so same-destination WMMAs are 4 instructions apart instead of 2, removing
// the VALU_DEP_2 stalls the round-1 asm showed on the accumulator RAW chain.
// ---------------------------------------------------------------------------

typedef float v2f __attribute__((ext_vector_type(2)));
typedef float v4f __attribute__((ext_vector_type(4)));
typedef float v8f __attribute__((ext_vector_type(8)));

#define HDIM 256
#define NTYPE 8
#define MT   32          // node rows per block tile (two 16-row WMMA subtiles)
#define BLK  512         // 16 wave32s per block
#define BPT  64          // blocks per type -> grid = 8*64 = 512 blocks
#define ASTR 260         // LDS stride (floats) for A tile: conflict-free b64 reads
#define NSTR 264         // LDS stride (floats) for nf2 tile
#define LN_EPS 1e-3f

// ws int layout: [0..7]=counts  [8..15]=cursors  [16..24]=offsets  [64..]=perm[N]
#define WS_CNT 0
#define WS_CUR 8
#define WS_OFF 16
#define WS_PERM 64

__global__ void hgt_ws_init(int* ws) {
    if (threadIdx.x < 32) ws[threadIdx.x] = 0;
}

__global__ void hgt_count(const int* __restrict__ tid, int* __restrict__ ws, int n) {
    int i = blockIdx.x * blockDim.x + threadIdx.x;
    if (i < n) atomicAdd(&ws[WS_CNT + tid[i]], 1);
}

__global__ void hgt_offsets(int* ws) {
    if (threadIdx.x == 0) {
        int acc = 0;
        for (int t = 0; t < NTYPE; ++t) { ws[WS_OFF + t] = acc; acc += ws[WS_CNT + t]; }
        ws[WS_OFF + NTYPE] = acc;
    }
}

__global__ void hgt_scatter(const int* __restrict__ tid, int* __restrict__ ws, int n) {
    int i = blockIdx.x * blockDim.x + threadIdx.x;
    if (i < n) {
        int t = tid[i];
        int pos = ws[WS_OFF + t] + atomicAdd(&ws[WS_CUR + t], 1);
        ws[WS_PERM + pos] = i;
    }
}

// ---------------------------------------------------------------------------
// Main fused kernel.
// ---------------------------------------------------------------------------
__global__ __launch_bounds__(BLK) void hgt_gemm_ln(
    const float* __restrict__ nf,
    const float* __restrict__ msg,
    const float* __restrict__ wa,
    const float* __restrict__ ba,
    const float* __restrict__ skip,
    const int*   __restrict__ ws,
    float*       __restrict__ out)
{
    __shared__ float ldsA[MT * ASTR];      // gathered msg tile  (33.3 KB)
    __shared__ float ldsO[MT * NSTR];      // gated nf2 tile     (33.8 KB)
    __shared__ float ldsStat[MT * 2];      // per-row mu, rsigma
    __shared__ int   ldsRows[MT];          // node ids of this tile

    const int t    = blockIdx.x / BPT;     // type handled by this block
    const int bin  = blockIdx.x % BPT;
    const int tid  = threadIdx.x;
    const int wid  = tid >> 5;             // wave id 0..15 -> output col slab
    const int lane = tid & 31;
    const int l15  = lane & 15;
    const int hf   = lane >> 4;            // lane half (A/B K-pair select)
    const int koff = hf * 2;

    const int cnt   = ws[WS_CNT + t];
    const int obase = ws[WS_OFF + t];
    const int ntile = (cnt + MT - 1) / MT;
    const int* perm = ws + WS_PERM;

    // per-type scalars
    const float alpha = 1.0f / (1.0f + expf(-skip[t]));
    const float oneMa = 1.0f - alpha;

    const int   colg = wid * 16 + l15;               // global output column
    const float bias = ba[t * HDIM + colg];

    // ---- Preload this wave's B slab: wa[t][:, colg] for all 256 K rows.
    // B 4x16 layout: lane=N col, VGPR pair = K rows {koff, koff+1} of each
    // 4-row chunk.  128 VGPRs, reused across every node tile of this type.
    const float* waT = wa + (size_t)t * HDIM * HDIM;
    v2f breg[64];
#pragma unroll
    for (int j = 0; j < 64; ++j) {
        int r = 4 * j + koff;
        breg[j].x = waT[(size_t)r * HDIM + colg];
        breg[j].y = waT[(size_t)(r + 1) * HDIM + colg];
    }

    for (int tile = bin; tile < ntile; tile += BPT) {
        const int rbase = tile * MT;
        if (tid < MT) {
            int rr = rbase + tid;
            ldsRows[tid] = (rr < cnt) ? perm[obase + rr] : -1;
        }
        __syncthreads();

        // ---- Cooperative gather of A tile: 32 rows x 256 f32 (b128 loads).
#pragma unroll
        for (int i = 0; i < (MT * HDIM / 4) / BLK; ++i) {
            int idx  = tid + i * BLK;          // float4 index
            int row  = idx >> 6;               // 64 float4 per row
            int c4   = (idx & 63) << 2;
            int node = ldsRows[row];
            v4f v = {0.f, 0.f, 0.f, 0.f};
            if (node >= 0) v = *(const v4f*)(msg + (size_t)node * HDIM + c4);
            *(v4f*)(ldsA + row * ASTR + c4) = v;
        }
        __syncthreads();

        // ---- 2 x 64 WMMA f32 16x16x4 with FOUR independent accumulator
        // chains (even/odd K) -> same-dest WMMAs are 4 instructions apart.
        v8f acc0a = {0, 0, 0, 0, 0, 0, 0, 0};
        v8f acc0b = {0, 0, 0, 0, 0, 0, 0, 0};
        v8f acc1a = {0, 0, 0, 0, 0, 0, 0, 0};
        v8f acc1b = {0, 0, 0, 0, 0, 0, 0, 0};
        const float* a0 = ldsA + l15 * ASTR + koff;
        const float* a1 = ldsA + (16 + l15) * ASTR + koff;
#pragma unroll
        for (int k = 0; k < 64; k += 2) {
            v2f av0e = *(const v2f*)(a0 + 4 * k);
            v2f av1e = *(const v2f*)(a1 + 4 * k);
            v2f av0o = *(const v2f*)(a0 + 4 * (k + 1));
            v2f av1o = *(const v2f*)(a1 + 4 * (k + 1));
            acc0a = __builtin_amdgcn_wmma_f32_16x16x4_f32(
                false, av0e, false, breg[k], (short)0, acc0a, false, false);
            acc1a = __builtin_amdgcn_wmma_f32_16x16x4_f32(
                false, av1e, false, breg[k], (short)0, acc1a, false, false);
            acc0b = __builtin_amdgcn_wmma_f32_16x16x4_f32(
                false, av0o, false, breg[k + 1], (short)0, acc0b, false, false);
            acc1b = __builtin_amdgcn_wmma_f32_16x16x4_f32(
                false, av1o, false, breg[k + 1], (short)0, acc1b, false, false);
        }
        v8f acc0 = acc0a + acc0b;
        v8f acc1 = acc1a + acc1b;

        // ---- Epilogue: bias + sigmoid-gated blend with nf; stage nf2 in LDS.
        float g0[8], g1[8];
#pragma unroll
        for (int v = 0; v < 8; ++v) {
            int r0 = v + hf * 8;               // C layout: row = vgpr + 8*half
            int r1 = r0 + 16;
            int n0 = ldsRows[r0], n1 = ldsRows[r1];
            float f0 = (n0 >= 0) ? nf[(size_t)n0 * HDIM + colg] : 0.f;
            float f1 = (n1 >= 0) ? nf[(size_t)n1 * HDIM + colg] : 0.f;
            g0[v] = f0 * oneMa + (acc0[v] + bias) * alpha;
            g1[v] = f1 * oneMa + (acc1[v] + bias) * alpha;
            ldsO[r0 * NSTR + colg] = g0[v];
            ldsO[r1 * NSTR + colg] = g1[v];
        }
        __syncthreads();

        // ---- LayerNorm stats: wave w reduces rows w and w+16 (8 vals/lane).
#pragma unroll
        for (int rr = 0; rr < 2; ++rr) {
            int row = wid + rr * 16;
            const float* p = ldsO + row * NSTR + lane * 8;
            float s1 = 0.f, s2 = 0.f;
#pragma unroll
            for (int j = 0; j < 8; ++j) { float x = p[j]; s1 += x; s2 += x * x; }
#pragma unroll
            for (int m = 16; m >= 1; m >>= 1) {
                s1 += __shfl_xor(s1, m, 32);
                s2 += __shfl_xor(s2, m, 32);
            }
            if (lane == 0) {
                float mu  = s1 * (1.0f / HDIM);
                float var = s2 * (1.0f / HDIM) - mu * mu;
                ldsStat[row * 2 + 0] = mu;
                ldsStat[row * 2 + 1] = rsqrtf(var + LN_EPS);
            }
        }
        __syncthreads();

        // ---- Normalize + scatter-store (coalesced 64B per lane half).
#pragma unroll
        for (int v = 0; v < 8; ++v) {
            int r0 = v + hf * 8;
            int r1 = r0 + 16;
            int n0 = ldsRows[r0], n1 = ldsRows[r1];
            if (n0 >= 0) {
                float mu = ldsStat[r0 * 2], rs = ldsStat[r0 * 2 + 1];
                out[(size_t)n0 * HDIM + colg] = (g0[v] - mu) * rs;
            }
            if (n1 >= 0) {
                float mu = ldsStat[r1 * 2], rs = ldsStat[r1 * 2 + 1];
                out[(size_t)n1 * HDIM + colg] = (g1[v] - mu) * rs;
            }
        }
        __syncthreads();   // protect ldsA/ldsRows before next iteration
    }
}

extern "C" void kernel_launch(void* const* d_in, const int* in_sizes, int n_in,
                              void* d_out, int out_size, void* d_ws, size_t ws_size,
                              hipStream_t stream) {
    const float* nf       = (const float*)d_in[0];
    // d_in[1] = nType one-hot (unused; nType_id carries the same info)
    const int*   nType_id = (const int*)d_in[2];
    const float* msg      = (const float*)d_in[3];
    const float* wa       = (const float*)d_in[4];
    const float* ba       = (const float*)d_in[5];
    const float* skip     = (const float*)d_in[6];
    float*       out      = (float*)d_out;
    int*         ws       = (int*)d_ws;

    const int N = in_sizes[2];   // node count

    hgt_ws_init<<<1, 32, 0, stream>>>(ws);
    hgt_count  <<<(N + 255) / 256, 256, 0, stream>>>(nType_id, ws, N);
    hgt_offsets<<<1, 1, 0, stream>>>(ws);
    hgt_scatter<<<(N + 255) / 256, 256, 0, stream>>>(nType_id, ws, N);
    hgt_gemm_ln<<<NTYPE * BPT, BLK, 0, stream>>>(nf, msg, wa, ba, skip, ws, out);
}